// SlidingWindowAttention_70231305225194
// MI455X (gfx1250) — compile-verified
//
#include <hip/hip_runtime.h>
#include <hip/hip_bf16.h>

// SlidingWindowAttention for MI455X (gfx1250), fp32 via V_WMMA_F32_16X16X4_F32.
// B=2, S=2048, D=1024, H=16, KVH=4, HD=64, WINDOW=64.

typedef __attribute__((ext_vector_type(2))) float v2f;
typedef __attribute__((ext_vector_type(8))) float v8f;

#define B_ 2
#define S_ 2048
#define D_ 1024
#define H_ 16
#define KVH_ 4
#define HD_ 64
#define WIN_ 64
#define M_ (B_ * S_)   // 4096 rows

// ---------------------------------------------------------------------------
// Register-blocked GEMM: each wave computes a 64x32 tile of C = A[M][K]@W[K][N]
// as a 4x2 grid of 16x16 WMMA tiles (8 v8f accumulators = 64 VGPRs -> no
// scratch spills; 4x4 blocking spilled past the 256-VGPR window).
// Per K-step of 4: 4 A-frags (b64) + 2 B-frags feed 8 WMMAs (~1 load/WMMA).
// A-frag: lane holds M=lane&15, K=(lane>>4)*2+{0,1}
// B-frag: lane holds K=(lane>>4)*2+{0,1}, N=lane&15
// C-frag: lane holds row=(lane>>4)*8+r, col=lane&15
// DO_ROPE: fused RoPE epilogue (even/odd pair partner via shfl_xor lane^1).
// ---------------------------------------------------------------------------
template <int DO_ROPE>
__global__ void gemm64x32_wmma(const float* __restrict__ A, const float* __restrict__ W,
                               float* __restrict__ C, int M, int N, int K,
                               const float* __restrict__ fcos,
                               const float* __restrict__ fsin) {
    const int lane  = threadIdx.x & 31;
    const int wave  = threadIdx.x >> 5;
    const int tileM = blockIdx.y * 64;
    const int tileN = (blockIdx.x * (blockDim.x >> 5) + wave) * 32;
    if (tileN >= N) return;  // wave-uniform

    const int l15  = lane & 15;
    const int koff = (lane >> 4) << 1;

    v8f acc[4][2] = {};

    const float* ap = A + (long)(tileM + l15) * K + koff;   // +4 per step
    const float* bp = W + (long)koff * N + (tileN + l15);   // +4*N per step
    const long aStride = 16L * K;                           // between M tiles (imm)

    for (int kk = 0; kk < K; kk += 4) {
        v2f a[4], b[2];
        #pragma unroll
        for (int i = 0; i < 4; ++i) {
            a[i].x = ap[i * aStride];
            a[i].y = ap[i * aStride + 1];
        }
        #pragma unroll
        for (int j = 0; j < 2; ++j) {
            b[j].x = bp[j * 16];
            b[j].y = bp[N + j * 16];
        }
        #pragma unroll
        for (int i = 0; i < 4; ++i)
            #pragma unroll
            for (int j = 0; j < 2; ++j)
                acc[i][j] = __builtin_amdgcn_wmma_f32_16x16x4_f32(
                    false, a[i], false, b[j], (short)0, acc[i][j], false, false);
        ap += 4;
        bp += (long)4 * N;
    }

    const int rowOff = (lane >> 4) << 3;
    #pragma unroll
    for (int i = 0; i < 4; ++i) {
        const int rowBase = tileM + i * 16 + rowOff;
        #pragma unroll
        for (int j = 0; j < 2; ++j) {
            const int col = tileN + j * 16 + l15;
            #pragma unroll
            for (int r = 0; r < 8; ++r) {
                float val = acc[i][j][r];
                if (DO_ROPE) {
                    // even/odd head-dim pair lives in adjacent lanes of the C fragment
                    const float partner = __shfl_xor(val, 1, 32);
                    const int pos = (rowBase + r) & (S_ - 1);   // rows are b*S + s
                    const int fi  = (col & (HD_ - 1)) >> 1;     // freq index in [0,32)
                    const float c = fcos[pos * (HD_ / 2) + fi];
                    const float s = fsin[pos * (HD_ / 2) + fi];
                    // even col: a=val,  b=partner -> a*c - b*s
                    // odd  col: a=partner, b=val  -> a*s + b*c
                    val = ((col & 1) == 0) ? (val * c - partner * s)
                                           : (partner * s + val * c);
                }
                C[(long)(rowBase + r) * N + col] = val;
            }
        }
    }
}

// ---------------------------------------------------------------------------
// Sliding-window attention: one wave handles (b, h, 16 query rows).
// Keys needed: [i0-64, i0+15] -> exactly 5 aligned 16-key tiles.
// Scores -> LDS p[16][83] (stride 83: coprime with 64 banks), softmax over
// 80 cols by lanes 0-15, then P(16x80) @ V(80x64) via WMMA.
// ---------------------------------------------------------------------------
__global__ void attn_wmma(const float* __restrict__ q, const float* __restrict__ kbuf,
                          const float* __restrict__ vbuf, float* __restrict__ o) {
    __shared__ float p[16][83];   // 80 columns used

    const int lane = threadIdx.x;
    const int i0   = blockIdx.x * 16;
    const int h    = blockIdx.y;
    const int b    = blockIdx.z;
    const int kvh  = h >> 2;                 // N_REP = H/KVH = 4

    const int l15  = lane & 15;
    const int koff = (lane >> 4) << 1;
    const int r8   = (lane >> 4) << 3;
    const int keystart = i0 - WIN_;

    // ---- scores: 5 key tiles --------------------------------------------
    const float* qrow = q + ((long)(b * S_ + i0 + l15) * H_ + h) * HD_;
    for (int t = 0; t < 5; ++t) {
        const int j0 = keystart + t * 16;    // multiple of 16 -> fully valid or fully <0
        if (j0 < 0) {
            #pragma unroll
            for (int r = 0; r < 8; ++r) p[r8 + r][t * 16 + l15] = -1e9f;
            continue;
        }
        const float* krow = kbuf + ((long)(b * S_ + j0 + l15) * KVH_ + kvh) * HD_;
        v8f acc = {};
        for (int kk = 0; kk < HD_; kk += 4) {
            const int kb = kk + koff;
            v2f a, bb;
            a.x  = qrow[kb];  a.y  = qrow[kb + 1];
            bb.x = krow[kb];  bb.y = krow[kb + 1];   // B[k][n] = K[key n][dim k]
            acc = __builtin_amdgcn_wmma_f32_16x16x4_f32(false, a, false, bb,
                                                        (short)0, acc, false, false);
        }
        #pragma unroll
        for (int r = 0; r < 8; ++r) {
            const int row  = i0 + r8 + r;
            const int colk = j0 + l15;
            float sc = fmaxf(acc[r] * 0.125f, -1e9f);   // 1/sqrt(64); clamp like reference
            if (colk > row || row - colk > WIN_) sc = -1e9f;
            p[r8 + r][t * 16 + l15] = sc;
        }
    }
    __syncthreads();

    // ---- row softmax over 80 columns (lanes 0-15, one row each) ---------
    if (lane < 16) {
        float m = -1e30f;
        for (int c = 0; c < 80; ++c) m = fmaxf(m, p[lane][c]);
        float sum = 0.0f;
        for (int c = 0; c < 80; ++c) {
            const float e = expf(p[lane][c] - m);   // -1e9 entries underflow to 0
            p[lane][c] = e;
            sum += e;
        }
        const float inv = 1.0f / sum;
        for (int c = 0; c < 80; ++c) p[lane][c] *= inv;
    }
    __syncthreads();

    // ---- O(16x64) = P(16x80) @ V(80x64) ---------------------------------
    for (int nt = 0; nt < 4; ++nt) {
        const int n0 = nt * 16;
        v8f acc = {};
        for (int kk = 0; kk < 80; kk += 4) {
            const int kb = kk + koff;
            v2f a, bb;
            a.x = p[l15][kb];
            a.y = p[l15][kb + 1];
            const int key0 = keystart + kb;
            const int key1 = key0 + 1;
            bb.x = (key0 >= 0)
                 ? vbuf[((long)(b * S_ + key0) * KVH_ + kvh) * HD_ + n0 + l15] : 0.0f;
            bb.y = (key1 >= 0)
                 ? vbuf[((long)(b * S_ + key1) * KVH_ + kvh) * HD_ + n0 + l15] : 0.0f;
            acc = __builtin_amdgcn_wmma_f32_16x16x4_f32(false, a, false, bb,
                                                        (short)0, acc, false, false);
        }
        #pragma unroll
        for (int r = 0; r < 8; ++r) {
            o[((long)(b * S_ + i0 + r8 + r) * H_ + h) * HD_ + n0 + l15] = acc[r];
        }
    }
}

// ---------------------------------------------------------------------------
extern "C" void kernel_launch(void* const* d_in, const int* in_sizes, int n_in,
                              void* d_out, int out_size, void* d_ws, size_t ws_size,
                              hipStream_t stream) {
    const float* x    = (const float*)d_in[0];
    const float* fcos = (const float*)d_in[1];
    const float* fsin = (const float*)d_in[2];
    const float* wq   = (const float*)d_in[3];
    const float* wk   = (const float*)d_in[4];
    const float* wv   = (const float*)d_in[5];
    const float* wo   = (const float*)d_in[6];
    float* out = (float*)d_out;

    // workspace: q[4096*1024] k[4096*256] v[4096*256] attn[4096*1024]  = 40 MiB
    float* qb = (float*)d_ws;
    float* kb = qb + (size_t)M_ * (H_ * HD_);
    float* vb = kb + (size_t)M_ * (KVH_ * HD_);
    float* at = vb + (size_t)M_ * (KVH_ * HD_);

    const dim3 blk(128);  // 4 waves, each computing a 64x32 tile -> 128 cols/block
    // Q = rope(x@wq), K = rope(x@wk), V = x@wv
    gemm64x32_wmma<1><<<dim3((H_ * HD_) / 128, M_ / 64), blk, 0, stream>>>(
        x, wq, qb, M_, H_ * HD_, D_, fcos, fsin);
    gemm64x32_wmma<1><<<dim3((KVH_ * HD_) / 128, M_ / 64), blk, 0, stream>>>(
        x, wk, kb, M_, KVH_ * HD_, D_, fcos, fsin);
    gemm64x32_wmma<0><<<dim3((KVH_ * HD_) / 128, M_ / 64), blk, 0, stream>>>(
        x, wv, vb, M_, KVH_ * HD_, D_, fcos, fsin);

    // attention: grid (query tiles, heads, batch), one wave per block
    attn_wmma<<<dim3(S_ / 16, H_, B_), dim3(32), 0, stream>>>(qb, kb, vb, at);

    // out = attn @ wo
    gemm64x32_wmma<0><<<dim3(D_ / 128, M_ / 64), blk, 0, stream>>>(
        at, wo, out, M_, D_, H_ * HD_, fcos, fsin);
}